// LocationAwareAttention_12197707120900
// MI455X (gfx1250) — compile-verified
//
#include <hip/hip_runtime.h>
#include <hip/hip_bf16.h>

// Location-aware attention, refactored: energy = enc · q with
// q[b,h,:] = Wpsi[h]^T @ phi[b,h,:]  (einsum re-association kills the 69-GFLOP
// psi_h intermediate; bpsi dropped because softmax is shift-invariant).
// enc-streaming GEMMs use V_WMMA_F32_16X16X4_F32; tiles are staged into LDS
// by the Tensor Data Mover (TENSOR_LOAD_TO_LDS + s_wait_tensorcnt), with an
// explicit s_wait_dscnt 0 before each DMA issue so the single-buffered tile
// is never overwritten while same-wave ds_loads are still outstanding.

#define B_   32
#define U_   4096
#define E_   512
#define DD_  512
#define H_   4
#define HD_  128

typedef float v2f __attribute__((ext_vector_type(2)));
typedef float v8f __attribute__((ext_vector_type(8)));
typedef unsigned int v4u __attribute__((ext_vector_type(4)));
typedef int v4i __attribute__((ext_vector_type(4)));
typedef int v8i __attribute__((ext_vector_type(8)));

#if defined(__has_builtin)
#if __has_builtin(__builtin_amdgcn_tensor_load_to_lds) && __has_builtin(__builtin_amdgcn_s_wait_tensorcnt)
#define HAVE_TDM 1
#endif
#endif

#ifdef HAVE_TDM
__device__ __forceinline__ void tdm_load_2d(unsigned int ldsOff, const void* gptr,
                                            const v8i& g1) {
    unsigned long long ga = (unsigned long long)gptr;
    v4u g0;
    g0.x = 1u;                                      // count=1, user descriptor
    g0.y = ldsOff;                                  // lds_addr (addr[31:0])
    g0.z = (unsigned int)ga;                        // global_addr[31:0]
    g0.w = (unsigned int)(ga >> 32) | (2u << 30);   // global_addr[56:32] | type=2
    v4i g2 = { 0, 0, 0, 0 };
    v4i g3 = { 0, 0, 0, 0 };
    // Drain pending same-wave LDS reads before the DMA may overwrite the tile.
    asm volatile("s_wait_dscnt 0" ::: "memory");
#if __clang_major__ >= 23
    __builtin_amdgcn_tensor_load_to_lds(g0, g1, g2, g3, (v8i){0,0,0,0,0,0,0,0}, 0);
#else
    __builtin_amdgcn_tensor_load_to_lds(g0, g1, g2, g3, 0);
#endif
    __builtin_amdgcn_s_wait_tensorcnt(0);
    __builtin_amdgcn_wave_barrier();
}
#endif

// ---------------- K1: phi[b, h*128+k] = dec[b,:] . Wphi[h,:,k] + bphi -------
__global__ void k_phi(const float* __restrict__ dec, const float* __restrict__ Wphi,
                      const float* __restrict__ bphi, float* __restrict__ phi) {
    int b = blockIdx.x;
    int t = threadIdx.x;            // t = h*128 + k
    int h = t >> 7, k = t & 127;
    const float* w = Wphi + (size_t)h * DD_ * HD_ + k;
    const float* x = dec + (size_t)b * DD_;
    float acc = bphi[t];
    for (int d = 0; d < DD_; ++d) acc += x[d] * w[(size_t)d * HD_];
    phi[(size_t)b * (H_ * HD_) + t] = acc;
}

// ---------------- K2: q[b,h,e] = sum_k phi[b,h,k] * Wpsi[h,e,k] -------------
__global__ void k_q(const float* __restrict__ phi, const float* __restrict__ Wpsi,
                    float* __restrict__ q) {
    int bh = blockIdx.x;            // b*4 + h
    int h = bh & 3;
    int e = threadIdx.x;
    __shared__ float sp[HD_];
    if (threadIdx.x < HD_) sp[threadIdx.x] = phi[(size_t)bh * HD_ + threadIdx.x];
    __syncthreads();
    const float* w = Wpsi + ((size_t)h * E_ + e) * HD_;
    float acc = 0.f;
    for (int k = 0; k < HD_; ++k) acc += sp[k] * w[k];
    q[(size_t)bh * E_ + e] = acc;
}

// ---------------- K3: conv31(SAME) + Wloc -> energy (init) ------------------
__global__ void k_loc(const float* __restrict__ prev, const float* __restrict__ ck,
                      const float* __restrict__ Wloc, float* __restrict__ energy) {
    int b = blockIdx.x;
    int u0 = blockIdx.y * 256;
    __shared__ float sp[256 + 30];
    __shared__ float sk[31 * 32];
    __shared__ float swl[32 * 4];
    int t = threadIdx.x;
    for (int i = t; i < 286; i += 256) {
        int u = u0 - 15 + i;
        sp[i] = (u >= 0 && u < U_) ? prev[(size_t)b * U_ + u] : 0.f;
    }
    for (int i = t; i < 992; i += 256) sk[i] = ck[i];
    if (t < 128) swl[t] = Wloc[t];
    __syncthreads();
    float f[32];
#pragma unroll
    for (int c = 0; c < 32; ++c) f[c] = 0.f;
    for (int tap = 0; tap < 31; ++tap) {
        float x = sp[t + tap];
#pragma unroll
        for (int c = 0; c < 32; ++c) f[c] += x * sk[tap * 32 + c];
    }
    int u = u0 + t;
#pragma unroll
    for (int h = 0; h < H_; ++h) {
        float acc = 0.f;
#pragma unroll
        for (int c = 0; c < 32; ++c) acc += f[c] * swl[c * 4 + h];
        energy[((size_t)b * H_ + h) * U_ + u] = acc;
    }
}

// ---------------- K4: energy += enc(16u x 512e) * q^T via WMMA f32 16x16x4 --
// A 16x4 frag: lanes0-15 hold K={0,1}, lanes16-31 K={2,3} (M = lane&15).
// B 4x16 frag: N = lane&15, VGPR0/1 hold K = 2*half + {0,1}; lanes with N>=4
// read a zero-filled LDS row through a precomputed pointer (no EXEC games).
__global__ void k_energy(const float* __restrict__ enc, const float* __restrict__ q,
                         float* __restrict__ energy) {
    int b = blockIdx.x;
    int wave = threadIdx.x >> 5, lane = threadIdx.x & 31;
    int u0 = blockIdx.y * 128 + wave * 16;
    __shared__ float sQ[5 * E_];             // rows 0-3 = q[b], row 4 = zeros
    __shared__ float sEnc[8][16 * 64];       // 32 KB (per-wave private tile)
    for (int i = threadIdx.x; i < H_ * E_; i += 256) sQ[i] = q[(size_t)b * H_ * E_ + i];
    for (int i = threadIdx.x; i < E_; i += 256) sQ[H_ * E_ + i] = 0.f;
    __syncthreads();
    int half = lane >> 4, m = lane & 15;
    const float* qrow = &sQ[(m < H_ ? m : H_) * E_];
    v8f c = {};
    const float* encB = enc + ((size_t)b * U_ + u0) * E_;
    float* se = &sEnc[wave][0];
#ifdef HAVE_TDM
    // D# (CDNA5 ISA): data_size=4B, tensor=512x4096,
    // tile=64x16, dim0_stride=512 -> packed 16x64 fp32 tile in LDS.
    const v8i g1 = { (int)0x00020000,
                     (int)(512u << 16),   // atomic_barrier_addr=0 | tensor_dim0[15:0]
                     (int)(4096u << 16),  // tensor_dim0[31:16]=0  | tensor_dim1[15:0]
                     (int)(64u << 16),    // tensor_dim1[31:16]=0  | tile_dim0=64
                     (int)16,             // tile_dim1=16 | tile_dim2=0
                     (int)512,            // tensor_dim0_stride[31:0]
                     0, 0 };
    unsigned int ldsOff = (unsigned int)(unsigned long long)se;  // addr[31:0]
#endif
    for (int ec = 0; ec < 8; ++ec) {        // 8 chunks of 64 e-columns
#ifdef HAVE_TDM
        tdm_load_2d(ldsOff, encB + ec * 64, g1);
#else
#pragma unroll
        for (int i = 0; i < 8; ++i) {       // coalesced stage: 16 rows x 64 cols
            int fi = lane + 32 * i;         // float4 index
            int row = fi >> 4, c4 = fi & 15;
            *(float4*)(se + row * 64 + c4 * 4) =
                *(const float4*)(encB + (size_t)row * E_ + ec * 64 + c4 * 4);
        }
        __builtin_amdgcn_wave_barrier();    // LDS is in-order within a wave
#endif
#pragma unroll
        for (int e4 = 0; e4 < 64; e4 += 4) {
            v2f a  = *(const v2f*)(se + m * 64 + e4 + 2 * half);
            v2f bb = *(const v2f*)(qrow + ec * 64 + e4 + 2 * half);
            c = __builtin_amdgcn_wmma_f32_16x16x4_f32(false, a, false, bb,
                                                      (short)0, c, false, false);
        }
        __builtin_amdgcn_wave_barrier();
    }
    // C/D: VGPR r -> M=r (lanes0-15) / M=r+8 (lanes16-31), N = lane&15 = head
    if (m < H_) {
        float* eb = energy + ((size_t)b * H_ + m) * U_ + u0;
#pragma unroll
        for (int r = 0; r < 8; ++r) eb[r + 8 * half] += c[r];
    }
}

// ---------------- K5: softmax over U, in place (energy -> attn) -------------
__global__ void k_softmax(float* __restrict__ energy) {
    float* e = energy + (size_t)blockIdx.x * U_;
    int t = threadIdx.x;
    __shared__ float red[256];
    float mx = -3.4e38f;
    for (int i = t; i < U_; i += 256) mx = fmaxf(mx, e[i]);
    red[t] = mx; __syncthreads();
    for (int s = 128; s > 0; s >>= 1) { if (t < s) red[t] = fmaxf(red[t], red[t + s]); __syncthreads(); }
    mx = red[0]; __syncthreads();
    float sum = 0.f;
    for (int i = t; i < U_; i += 256) { float v = __expf(e[i] - mx); e[i] = v; sum += v; }
    red[t] = sum; __syncthreads();
    for (int s = 128; s > 0; s >>= 1) { if (t < s) red[t] += red[t + s]; __syncthreads(); }
    float inv = 1.0f / red[0];
    for (int i = t; i < U_; i += 256) e[i] *= inv;
}

// ---------------- K6: ctx[b,h,e] = sum_u attn[b,h,u]*enc[b,u,e] via WMMA ----
// Per block: one 16-wide e-stripe; 8 waves split K=U, partials reduced in LDS.
// attn chunks (4x128 tile of the 4x4096 attn matrix) staged via TDM; A-frag
// rows with M>=4 read a per-wave zero row through a precomputed pointer.
__global__ void k_context(const float* __restrict__ enc, const float* __restrict__ attn,
                          float* __restrict__ ctx) {
    int b = blockIdx.x;
    int e0 = blockIdx.y * 16;
    int wave = threadIdx.x >> 5, lane = threadIdx.x & 31;
    int half = lane >> 4, m = lane & 15;
    __shared__ float sAttn[8][5 * 128];      // rows 0-3 = attn chunk, row 4 = zeros
    __shared__ float sC[8][H_][16];
    for (int i = lane; i < 128; i += 32) sAttn[wave][4 * 128 + i] = 0.f;
    __builtin_amdgcn_wave_barrier();
    const float* arow = &sAttn[wave][(m < H_ ? m : H_) * 128];
    v8f c = {};
    const float* encB = enc + (size_t)b * U_ * E_ + e0;
    const float* attnB = attn + (size_t)b * H_ * U_;
    int uW = wave * 512;
#ifdef HAVE_TDM
    // D#: data_size=4B, tensor=4096x4, tile=128x4, dim0_stride=4096.
    const v8i g1a = { (int)0x00020000,
                      (int)(4096u << 16),  // tensor_dim0[15:0] = 4096
                      (int)(4u << 16),     // tensor_dim1[15:0] = 4
                      (int)(128u << 16),   // tile_dim0 = 128
                      (int)4,              // tile_dim1 = 4
                      (int)4096,           // tensor_dim0_stride
                      0, 0 };
    unsigned int ldsOffA = (unsigned int)(unsigned long long)&sAttn[wave][0];
#endif
    for (int sc = 0; sc < 4; ++sc) {
        int uBase = uW + sc * 128;
#ifdef HAVE_TDM
        tdm_load_2d(ldsOffA, attnB + uBase, g1a);
#else
#pragma unroll
        for (int i = 0; i < 4; ++i) {        // stage attn[0..3][uBase..+128)
            int fi = lane + 32 * i;          // 128 float4s
            int row = fi >> 5, c4 = fi & 31;
            *(float4*)(&sAttn[wave][row * 128 + c4 * 4]) =
                *(const float4*)(attnB + (size_t)row * U_ + uBase + c4 * 4);
        }
        __builtin_amdgcn_wave_barrier();
#endif
#pragma unroll
        for (int k4 = 0; k4 < 128; k4 += 4) {
            int u = uBase + k4 + 2 * half;
            v2f a = *(const v2f*)(arow + k4 + 2 * half);
            v2f bb;
            bb.x = encB[(size_t)u * E_ + m];
            bb.y = encB[(size_t)(u + 1) * E_ + m];
            c = __builtin_amdgcn_wmma_f32_16x16x4_f32(false, a, false, bb,
                                                      (short)0, c, false, false);
        }
        __builtin_amdgcn_wave_barrier();
    }
    if (half == 0) {                         // rows 0..3 live in c[0..3], lanes 0-15
#pragma unroll
        for (int r = 0; r < H_; ++r) sC[wave][r][m] = c[r];
    }
    __syncthreads();
    if (threadIdx.x < 64) {
        int r = threadIdx.x >> 4, n = threadIdx.x & 15;
        float acc = 0.f;
#pragma unroll
        for (int w = 0; w < 8; ++w) acc += sC[w][r][n];
        ctx[((size_t)b * H_ + r) * E_ + e0 + n] = acc;
    }
}

// ---------------- K7: attention_weights[b,u] = mean over heads --------------
__global__ void k_mean(const float* __restrict__ attn, float* __restrict__ aw) {
    int idx = blockIdx.x * 256 + threadIdx.x;     // b*U + u
    int b = idx >> 12, u = idx & (U_ - 1);
    const float* a = attn + (size_t)b * H_ * U_ + u;
    aw[idx] = 0.25f * (a[0] + a[U_] + a[2 * U_] + a[3 * U_]);
}

// ---------------- K8: out = ctx_flat @ Wout + bout --------------------------
__global__ void k_out(const float* __restrict__ ctx, const float* __restrict__ Wout,
                      const float* __restrict__ bout, float* __restrict__ out) {
    int b = blockIdx.x, a = threadIdx.x;
    __shared__ float sc[H_ * E_];
    for (int i = threadIdx.x; i < H_ * E_; i += 512) sc[i] = ctx[(size_t)b * H_ * E_ + i];
    __syncthreads();
    float acc = bout[a];
    for (int i = 0; i < H_ * E_; ++i) acc += sc[i] * Wout[(size_t)i * 512 + a];
    out[(size_t)b * 512 + a] = acc;
}

extern "C" void kernel_launch(void* const* d_in, const int* in_sizes, int n_in,
                              void* d_out, int out_size, void* d_ws, size_t ws_size,
                              hipStream_t stream) {
    const float* dec   = (const float*)d_in[0];
    const float* enc   = (const float*)d_in[1];
    const float* prev  = (const float*)d_in[2];
    const float* Wphi  = (const float*)d_in[3];
    const float* bphi  = (const float*)d_in[4];
    const float* Wpsi  = (const float*)d_in[5];
    // d_in[6] = bpsi: adds a per-(b,h) constant to energy -> softmax-invariant, unused
    const float* ckern = (const float*)d_in[7];
    const float* Wloc  = (const float*)d_in[8];
    const float* Wout  = (const float*)d_in[9];
    const float* bout  = (const float*)d_in[10];

    float* ws     = (float*)d_ws;
    float* phi    = ws;                       // 32*512        = 16384
    float* q      = phi + 16384;              // 32*4*512      = 65536
    float* energy = q + 65536;                // 32*4*4096     = 524288 (attn in place)
    float* ctx    = energy + 524288;          // 32*4*512      = 65536

    float* out_ctx = (float*)d_out;           // B*512
    float* out_aw  = out_ctx + B_ * 512;      // B*U

    k_phi    <<<dim3(B_),        dim3(512), 0, stream>>>(dec, Wphi, bphi, phi);
    k_q      <<<dim3(B_ * H_),   dim3(512), 0, stream>>>(phi, Wpsi, q);
    k_loc    <<<dim3(B_, 16),    dim3(256), 0, stream>>>(prev, ckern, Wloc, energy);
    k_energy <<<dim3(B_, 32),    dim3(256), 0, stream>>>(enc, q, energy);
    k_softmax<<<dim3(B_ * H_),   dim3(256), 0, stream>>>(energy);
    k_context<<<dim3(B_, 32),    dim3(256), 0, stream>>>(enc, energy, ctx);
    k_mean   <<<dim3(512),       dim3(256), 0, stream>>>(energy, out_aw);
    k_out    <<<dim3(B_),        dim3(512), 0, stream>>>(ctx, Wout, bout, out_ctx);
}